// KandK_25022479466585
// MI455X (gfx1250) — compile-verified
//
#include <hip/hip_runtime.h>

typedef __attribute__((ext_vector_type(2))) float v2f;
typedef __attribute__((ext_vector_type(4))) float v4f;
typedef __attribute__((ext_vector_type(8))) float v8f;

#define NMAT 128
#define LDW  132   // padded LDS row stride -> conflict-free strided access
#define NPOW 21    // stored powers: F^1..F^12, F^24,F^36,...,F^120

// ---------------------------------------------------------------------------
// One 128x128x128 f32 matmul step on a whole workgroup (8 waves x 16-row strip)
// using V_WMMA_F32_16X16X4_F32. A row-major in LDS, B transposed in LDS,
// D written transposed (so it can be the next step's B).
// ---------------------------------------------------------------------------
__device__ __forceinline__ void wgp_matmul_step(const float* __restrict__ Ab,
                                                const float* Fcur, float* Gbuf,
                                                int wave, int r, int half)
{
    v8f acc[8];
    #pragma unroll
    for (int tn = 0; tn < 8; ++tn) acc[tn] = (v8f){0.f,0.f,0.f,0.f,0.f,0.f,0.f,0.f};

    // A fragment: lane holds A[m][k+2*half .. +1], m = wave*16 + r
    const float* ar = &Ab[(wave * 16 + r) * LDW + 2 * half];

    #pragma unroll 4
    for (int k4 = 0; k4 < 32; ++k4) {
        const int k = k4 * 4;
        v2f a = *(const v2f*)(ar + k);
        #pragma unroll
        for (int tn = 0; tn < 8; ++tn) {
            v2f bb = *(const v2f*)(&Fcur[(tn * 16 + r) * LDW + k + 2 * half]);
            acc[tn] = __builtin_amdgcn_wmma_f32_16x16x4_f32(
                false, a, false, bb, (short)0, acc[tn], false, false);
        }
    }
    // Store D transposed: lane holds rows (wave*16 + v + 8*half), col = tn*16 + r
    #pragma unroll
    for (int tn = 0; tn < 8; ++tn) {
        float* g = &Gbuf[(tn * 16 + r) * LDW + wave * 16 + half * 8];
        v4f lo = { acc[tn][0], acc[tn][1], acc[tn][2], acc[tn][3] };
        v4f hi = { acc[tn][4], acc[tn][5], acc[tn][6], acc[tn][7] };
        *(v4f*)(g)     = lo;
        *(v4f*)(g + 4) = hi;
    }
}

// Block-wide trace of LDS matrix (diagonal invariant under transpose) -> ln|tr|
__device__ __forceinline__ void block_trace_ln(const float* F, float* red, int t,
                                               float* out_ptr)
{
    if (t < 128) red[t] = F[t * LDW + t];
    __syncthreads();
    if (t < 64) red[t] += red[t + 64];
    __syncthreads();
    if (t < 16) red[t] += red[t + 16] + red[t + 32] + red[t + 48];
    __syncthreads();
    if (t == 0) {
        float tr = 0.f;
        #pragma unroll
        for (int i = 0; i < 16; ++i) tr += red[i];
        *out_ptr = logf(fabsf(tr));
    }
    __syncthreads();
}

// Coalesced dense copy of a padded LDS matrix to global (keeps orientation).
__device__ __forceinline__ void copy_lds_to_global(const float* Fsrc,
                                                   float* __restrict__ dst, int t)
{
    #pragma unroll
    for (int i = 0; i < 16; ++i) {
        int e = (i * 256 + t) * 4;
        int c = e >> 7, rr = e & 127;
        v4f v = *(const v4f*)(&Fsrc[c * LDW + rr]);
        *(v4f*)(&dst[e]) = v;
    }
}

__device__ __forceinline__ void load_x_to_lds(const float* __restrict__ xb,
                                              float* Ab, float* Fa, int t)
{
    #pragma unroll
    for (int i = 0; i < 16; ++i) {
        int idx4 = i * 256 + t;
        int e    = idx4 * 4;
        v4f v    = *(const v4f*)(xb + e);
        int row  = e >> 7;
        int col  = e & 127;
        *(v4f*)(&Ab[row * LDW + col]) = v;
        Fa[(col + 0) * LDW + row] = v[0];
        Fa[(col + 1) * LDW + row] = v[1];
        Fa[(col + 2) * LDW + row] = v[2];
        Fa[(col + 3) * LDW + row] = v[3];
    }
}

// ---------------------------------------------------------------------------
// FAST PATH phase A: depth-20 chain. Computes F^2..F^12 (traces 1..12 direct),
// then F^24..F^120 by multiplying with F^12. All 21 powers stored to global
// (transposed orientation; the pair-trace bilinear form is transpose-invariant).
// ---------------------------------------------------------------------------
__launch_bounds__(256, 1)
__global__ void trace_chain_fast(const float* __restrict__ x,
                                 float* __restrict__ lnabs,
                                 float* __restrict__ pw)
{
    __shared__ float Ab[NMAT * LDW];
    __shared__ float Fa[NMAT * LDW];
    __shared__ float Fb[NMAT * LDW];
    __shared__ float red[128];

    const int t    = threadIdx.x;
    const int b    = blockIdx.x;
    const int lane = t & 31;
    const int wave = t >> 5;
    const int r    = lane & 15;
    const int half = lane >> 4;

    float* base = pw + (size_t)b * NPOW * (NMAT * NMAT);

    load_x_to_lds(x + (size_t)b * NMAT * NMAT, Ab, Fa, t);
    __syncthreads();

    copy_lds_to_global(Fa, base, t);                 // F^1
    block_trace_ln(Fa, red, t, &lnabs[b * NMAT + 0]);

    float* Fcur = Fa;
    float* Gbuf = Fb;

    // F^2 .. F^12 : A = x (row-major), chain on B side
    for (int s = 2; s <= 12; ++s) {
        wgp_matmul_step(Ab, Fcur, Gbuf, wave, r, half);
        __syncthreads();
        { float* tmp = Fcur; Fcur = Gbuf; Gbuf = tmp; }
        copy_lds_to_global(Fcur, base + (size_t)(s - 1) * (NMAT * NMAT), t);
        block_trace_ln(Fcur, red, t, &lnabs[b * NMAT + (s - 1)]);
    }

    // Transpose F^12 (currently transposed in Fcur) into Ab as the new A operand
    #pragma unroll
    for (int i = 0; i < 64; ++i) {
        int e = i * 256 + t;
        int row = e >> 7, col = e & 127;
        Ab[row * LDW + col] = Fcur[col * LDW + row];
    }
    __syncthreads();

    // F^24 .. F^120 : F^{12k} = F^12 * F^{12(k-1)} (powers commute)
    for (int k = 2; k <= 10; ++k) {
        wgp_matmul_step(Ab, Fcur, Gbuf, wave, r, half);
        __syncthreads();
        { float* tmp = Fcur; Fcur = Gbuf; Gbuf = tmp; }
        copy_lds_to_global(Fcur, base + (size_t)(10 + k) * (NMAT * NMAT), t);
        __syncthreads();
    }
}

// ---------------------------------------------------------------------------
// FAST PATH phase B: tr(F^i) = tr(F^{12k} * F^r), i = 12k + r, r in 1..12.
// One block per (batch, i); 7424 fully parallel blocks, all reads L2-resident.
// Bilinear form sum_{p,q} A[p,q] * B[q,p] is invariant under (A,B)->(A^T,B^T).
// ---------------------------------------------------------------------------
__global__ void trace_pairs_kernel(const float* __restrict__ pw,
                                   float* __restrict__ lnabs)
{
    const int b  = blockIdx.x;
    const int i  = 13 + blockIdx.y;             // 13..128
    const int r  = ((i - 1) % 12) + 1;          // 1..12
    const int k  = (i - r) / 12;                // 1..10
    const int pA = (k == 1) ? 11 : (10 + k);    // F^{12k}
    const int pB = r - 1;                       // F^r

    const float* __restrict__ A = pw + ((size_t)b * NPOW + pA) * (NMAT * NMAT);
    const float* __restrict__ B = pw + ((size_t)b * NPOW + pB) * (NMAT * NMAT);

    const int t  = threadIdx.x;                 // 256 threads
    const int i0 = t >> 1;                      // row of A, col of B
    const int hf = t & 1;

    const float* arow = A + i0 * NMAT + hf * 64;
    const float* bcol = B + hf * 64 * NMAT + i0;

    float s = 0.f;
    #pragma unroll 4
    for (int j4 = 0; j4 < 16; ++j4) {
        v4f a = *(const v4f*)(arow + j4 * 4);
        float b0 = bcol[(j4 * 4 + 0) * NMAT];
        float b1 = bcol[(j4 * 4 + 1) * NMAT];
        float b2 = bcol[(j4 * 4 + 2) * NMAT];
        float b3 = bcol[(j4 * 4 + 3) * NMAT];
        s += a[0] * b0 + a[1] * b1 + a[2] * b2 + a[3] * b3;
    }

    __shared__ float red[256];
    red[t] = s;
    __syncthreads();
    for (int off = 128; off > 0; off >>= 1) {
        if (t < off) red[t] += red[t + off];
        __syncthreads();
    }
    if (t == 0) lnabs[b * NMAT + (i - 1)] = logf(fabsf(red[0]));
}

// ---------------------------------------------------------------------------
// FALLBACK: full 127-step sequential chain (used if workspace is too small).
// ---------------------------------------------------------------------------
__launch_bounds__(256, 1)
__global__ void trace_chain_full(const float* __restrict__ x,
                                 float* __restrict__ lnabs)
{
    __shared__ float Ab[NMAT * LDW];
    __shared__ float Fa[NMAT * LDW];
    __shared__ float Fb[NMAT * LDW];
    __shared__ float red[128];

    const int t    = threadIdx.x;
    const int b    = blockIdx.x;
    const int lane = t & 31;
    const int wave = t >> 5;
    const int r    = lane & 15;
    const int half = lane >> 4;

    load_x_to_lds(x + (size_t)b * NMAT * NMAT, Ab, Fa, t);
    __syncthreads();

    float* Fcur = Fa;
    float* Gbuf = Fb;

    block_trace_ln(Fcur, red, t, &lnabs[b * NMAT + 0]);
    for (int s = 1; s < NMAT; ++s) {
        wgp_matmul_step(Ab, Fcur, Gbuf, wave, r, half);
        __syncthreads();
        { float* tmp = Fcur; Fcur = Gbuf; Gbuf = tmp; }
        block_trace_ln(Fcur, red, t, &lnabs[b * NMAT + s]);
    }
}

// ---------------------------------------------------------------------------
// Tail MLP: h = exp(W1 @ lnabs + b1); s = softmax_k(h); c[b,j] = sum s*W2[l,j,k]
// ---------------------------------------------------------------------------
__global__ void mlp_tail_kernel(const float* __restrict__ lnabs,
                                const float* __restrict__ W1,
                                const float* __restrict__ b1,
                                const float* __restrict__ W2,
                                float* __restrict__ out)
{
    __shared__ float ln[128];
    __shared__ float sft[4][64];
    __shared__ float mx[4], sm[4];

    const int b = blockIdx.x;
    const int t = threadIdx.x;   // 0..63

    ln[t]      = lnabs[b * 128 + t];
    ln[t + 64] = lnabs[b * 128 + 64 + t];
    __syncthreads();

    #pragma unroll
    for (int l = 0; l < 4; ++l) {
        const float* __restrict__ w = &W1[(l * 64 + t) * 128];
        float acc = b1[l * 64 + t];
        #pragma unroll 8
        for (int n = 0; n < 128; ++n) acc += ln[n] * w[n];
        sft[l][t] = expf(acc);
    }
    __syncthreads();

    if (t < 4) {
        float m = sft[t][0];
        for (int kk = 1; kk < 64; ++kk) m = fmaxf(m, sft[t][kk]);
        float z = 0.f;
        for (int kk = 0; kk < 64; ++kk) z += expf(sft[t][kk] - m);
        mx[t] = m; sm[t] = z;
    }
    __syncthreads();

    #pragma unroll
    for (int l = 0; l < 4; ++l)
        sft[l][t] = expf(sft[l][t] - mx[l]) / sm[l];
    __syncthreads();

    float c = 0.f;
    #pragma unroll
    for (int l = 0; l < 4; ++l) {
        const float* __restrict__ w2 = &W2[(l * 64 + t) * 64];
        #pragma unroll 8
        for (int kk = 0; kk < 64; ++kk) c += sft[l][kk] * w2[kk];
    }
    out[b * 64 + t] = c;
}

extern "C" void kernel_launch(void* const* d_in, const int* in_sizes, int n_in,
                              void* d_out, int out_size, void* d_ws, size_t ws_size,
                              hipStream_t stream)
{
    (void)in_sizes; (void)n_in; (void)out_size;
    const float* x  = (const float*)d_in[0];   // [64,128,128]
    const float* W1 = (const float*)d_in[1];   // [4,64,128]
    const float* b1 = (const float*)d_in[2];   // [4,64]
    const float* W2 = (const float*)d_in[3];   // [4,64,64]
    float* out   = (float*)d_out;              // [64,64]
    float* lnabs = (float*)d_ws;               // [64,128]
    float* pw    = lnabs + 64 * 128;           // [64,21,128,128] power matrices

    const size_t need = (size_t)64 * 128 * sizeof(float)
                      + (size_t)64 * NPOW * NMAT * NMAT * sizeof(float); // ~88 MB

    if (ws_size >= need) {
        trace_chain_fast<<<64, 256, 0, stream>>>(x, lnabs, pw);
        trace_pairs_kernel<<<dim3(64, 116), 256, 0, stream>>>(pw, lnabs);
    } else {
        trace_chain_full<<<64, 256, 0, stream>>>(x, lnabs);
    }
    mlp_tail_kernel<<<64, 64, 0, stream>>>(lnabs, W1, b1, W2, out);
}